// CAE_78469052498044
// MI455X (gfx1250) — compile-verified
//
#include <hip/hip_runtime.h>
#include <hip/hip_bf16.h>
#include <math.h>

typedef float v2f __attribute__((ext_vector_type(2)));
typedef float v8f __attribute__((ext_vector_type(8)));

#define BATCH 64

// ---------------------------------------------------------------------------
// Layer 0 only (Cin=1): fused direct conv3x3(SAME) + ReLU + 2x2 maxpool.
// ---------------------------------------------------------------------------
template <int CIN, int COUT, int HIN>
__global__ void conv_relu_pool(const float* __restrict__ in,
                               const float* __restrict__ w,
                               const float* __restrict__ bias,
                               float* __restrict__ out) {
  constexpr int HOUT = HIN / 2;
  int idx = blockIdx.x * blockDim.x + threadIdx.x;
  constexpr int TOTAL = BATCH * COUT * HOUT * HOUT;
  if (idx >= TOTAL) return;
  int ox = idx % HOUT;
  int oy = (idx / HOUT) % HOUT;
  int co = (idx / (HOUT * HOUT)) % COUT;
  int b  = idx / (HOUT * HOUT * COUT);

  int iy0 = oy * 2 - 1;
  int ix0 = ox * 2 - 1;

  float acc0 = 0.f, acc1 = 0.f, acc2 = 0.f, acc3 = 0.f;
  const float* wbase = w + (size_t)co * CIN * 9;
  const float* inb = in + (size_t)b * CIN * HIN * HIN;

  for (int ci = 0; ci < CIN; ++ci) {
    const float* ip = inb + (size_t)ci * HIN * HIN;
    float patch[4][4];
#pragma unroll
    for (int r = 0; r < 4; ++r) {
      int yy = iy0 + r;
      bool yok = (yy >= 0) && (yy < HIN);
#pragma unroll
      for (int cc = 0; cc < 4; ++cc) {
        int xx = ix0 + cc;
        bool ok = yok && (xx >= 0) && (xx < HIN);
        patch[r][cc] = ok ? ip[yy * HIN + xx] : 0.0f;
      }
    }
    float wv[9];
#pragma unroll
    for (int t = 0; t < 9; ++t) wv[t] = wbase[ci * 9 + t];
#pragma unroll
    for (int py = 0; py < 2; ++py) {
#pragma unroll
      for (int px = 0; px < 2; ++px) {
        float s = 0.f;
#pragma unroll
        for (int ky = 0; ky < 3; ++ky)
#pragma unroll
          for (int kx = 0; kx < 3; ++kx)
            s = fmaf(wv[ky * 3 + kx], patch[py + ky][px + kx], s);
        if (py == 0 && px == 0) acc0 += s;
        else if (py == 0 && px == 1) acc1 += s;
        else if (py == 1 && px == 0) acc2 += s;
        else acc3 += s;
      }
    }
  }
  float m = fmaxf(fmaxf(acc0, acc1), fmaxf(acc2, acc3));
  out[idx] = fmaxf(m + bias[co], 0.0f);
}

// ---------------------------------------------------------------------------
// LDS gather helper for implicit-GEMM A fragments.
// k = ci*9 + ky*3 + kx; addr = ci*S2 + base + ky*W2 + kx (padded layout).
// ---------------------------------------------------------------------------
__device__ __forceinline__ float lds_at(const float* lds, int k, int base,
                                        int S2, int W2) {
  int ci = k / 9;
  int r  = k - ci * 9;
  int ky = r / 3;
  int kx = r - ky * 3;
  return lds[ci * S2 + base + ky * W2 + kx];
}

// ---------------------------------------------------------------------------
// Implicit-GEMM conv3x3(SAME) via V_WMMA_F32_16X16X4_F32.
// One workgroup (8 waves) per (sample, RS-row segment): stage the zero-padded
// input segment [CIN x (RS+2) x (HIN+2)] into LDS, then each wave produces
// 16(pos) x 16(cout) output tiles with a K-loop over CIN*9.
// Output layout: go[b][pos][cout] (pre-pool, pre-bias).
// ---------------------------------------------------------------------------
template <int CIN, int COUT, int HIN, int RS>
__global__ void __launch_bounds__(256)
conv_wmma(const float* __restrict__ in, const float* __restrict__ w,
          float* __restrict__ go) {
  constexpr int W2 = HIN + 2;
  constexpr int S2 = (RS + 2) * W2;
  constexpr int LDSZ = CIN * S2;
  constexpr int K = CIN * 9;
  constexpr int PSEG = RS * HIN;      // positions in this segment
  constexpr int NT_M = PSEG / 16;
  constexpr int NT_N = COUT / 16;
  constexpr int T = NT_M * NT_N;      // WMMA tiles per workgroup
  constexpr int SEGS = HIN / RS;
  __shared__ float lds[LDSZ];

  int b   = blockIdx.x / SEGS;
  int seg = blockIdx.x % SEGS;
  int y0  = seg * RS;
  const float* inb = in + (size_t)b * CIN * HIN * HIN;

  // Stage padded input segment into LDS (zero borders).
  for (int u = threadIdx.x; u < LDSZ; u += 256) {
    int ci  = u / S2;
    int rem = u - ci * S2;
    int rr  = rem / W2;         // 0..RS+1
    int cc  = rem - rr * W2;    // 0..HIN+1
    int y = y0 + rr - 1;
    int x = cc - 1;
    float v = 0.0f;
    if (y >= 0 && y < HIN && x >= 0 && x < HIN)
      v = inb[(ci * HIN + y) * HIN + x];
    lds[u] = v;
  }
  __syncthreads();

  int wid  = threadIdx.x >> 5;
  int lane = threadIdx.x & 31;
  int half = lane >> 4;
  int l    = lane & 15;

  for (int t = wid; t < T; t += 8) {
    int tm = t % NT_M, tn = t / NT_M;
    int m  = tm * 16 + l;              // A row: local position
    int ly = m / HIN, x = m % HIN;
    int abase = ly * W2 + x;
    const float* bp = w + (size_t)(tn * 16 + l) * K + half * 2;

    v8f c = {};
#pragma unroll 4
    for (int kk = 0; kk < K; kk += 4) {
      int k0 = kk + half * 2;
      v2f a = {lds_at(lds, k0, abase, S2, W2),
               lds_at(lds, k0 + 1, abase, S2, W2)};
      v2f bv = *(const v2f*)(bp + kk);
      c = __builtin_amdgcn_wmma_f32_16x16x4_f32(
          false, a, false, bv, (short)0, c, false, false);
    }
    int pbase = b * (HIN * HIN) + y0 * HIN + tm * 16;
#pragma unroll
    for (int r = 0; r < 8; ++r) {
      int p = pbase + r + half * 8;
      go[(size_t)p * COUT + tn * 16 + l] = c[r];
    }
  }
}

// ---------------------------------------------------------------------------
// Pool + bias + ReLU + transpose: go[b][pos][co] -> out[b][co][oy][ox]
// ---------------------------------------------------------------------------
template <int COUT, int HIN>
__global__ void pool_bias_relu(const float* __restrict__ go,
                               const float* __restrict__ bias,
                               float* __restrict__ out) {
  constexpr int HOUT = HIN / 2, P = HIN * HIN;
  int idx = blockIdx.x * blockDim.x + threadIdx.x;
  constexpr int TOTAL = BATCH * COUT * HOUT * HOUT;
  if (idx >= TOTAL) return;
  int ox = idx % HOUT;
  int oy = (idx / HOUT) % HOUT;
  int co = (idx / (HOUT * HOUT)) % COUT;
  int b  = idx / (HOUT * HOUT * COUT);
  const float* g = go + (size_t)b * P * COUT + co;
  int p0 = (2 * oy) * HIN + 2 * ox;
  float v0 = g[(size_t)p0 * COUT];
  float v1 = g[(size_t)(p0 + 1) * COUT];
  float v2 = g[(size_t)(p0 + HIN) * COUT];
  float v3 = g[(size_t)(p0 + HIN + 1) * COUT];
  float m = fmaxf(fmaxf(v0, v1), fmaxf(v2, v3));
  out[idx] = fmaxf(m + bias[co], 0.0f);
}

// ---------------------------------------------------------------------------
// FC1: h2 = relu(h[64,2048] @ wl1^T + bl1) via V_WMMA_F32_16X16X4_F32.
// ---------------------------------------------------------------------------
__global__ void __launch_bounds__(32)
fc1_wmma(const float* __restrict__ h, const float* __restrict__ wl1,
         const float* __restrict__ bl1, float* __restrict__ h2) {
  int tile = blockIdx.x;          // 64 tiles: 4 (M) x 16 (N)
  int tm = tile >> 4;
  int tn = tile & 15;
  int lane = threadIdx.x;
  int half = lane >> 4;
  int l = lane & 15;

  const float* ap = h   + (size_t)(tm * 16 + l) * 2048 + half * 2;
  const float* bp = wl1 + (size_t)(tn * 16 + l) * 2048 + half * 2;

  v8f c = {};
#pragma unroll 8
  for (int k = 0; k < 2048; k += 4) {
    v2f a = *(const v2f*)(ap + k);
    v2f b = *(const v2f*)(bp + k);
    c = __builtin_amdgcn_wmma_f32_16x16x4_f32(
        false, a, false, b, (short)0, c, false, false);
  }
#pragma unroll
  for (int r = 0; r < 8; ++r) {
    int m = tm * 16 + r + half * 8;
    int n = tn * 16 + l;
    h2[m * 256 + n] = fmaxf(c[r] + bl1[n], 0.0f);
  }
}

// ---------------------------------------------------------------------------
// FC2 + hardtanh + per-sample derived constants; writes X_ANGLE output tail.
// ---------------------------------------------------------------------------
__global__ void fc2_derive(const float* __restrict__ h2,
                           const float* __restrict__ wl2,
                           const float* __restrict__ bl2,
                           const float* __restrict__ pos,
                           float* __restrict__ derived,
                           float* __restrict__ xang_out) {
  int b = blockIdx.x * blockDim.x + threadIdx.x;
  if (b >= BATCH) return;
  const float* hv = h2 + (size_t)b * 256;
  float p0 = bl2[0], p1 = bl2[1];
  for (int n = 0; n < 256; ++n) {
    float hh = hv[n];
    p0 = fmaf(hh, wl2[n], p0);
    p1 = fmaf(hh, wl2[256 + n], p1);
  }
  p0 = fminf(fmaxf(p0, -1.0f), 1.0f);
  p1 = fminf(fmaxf(p1, -1.0f), 1.0f);

  const float PI = 3.14159265358979323846f;
  float XA  = (p0 + 1.0f) * (PI * 0.5f - PI / 18.0f) * 0.5f + PI / 18.0f;
  float SL  = ((p1 + 1.0f) * (0.35f - 0.1f) * 0.5f + 0.1f) * 32.0f;
  float SLZ = SL * 0.2f;

  float theta = -pos[b] + PI * 0.5f;
  float ct = cosf(theta), st = sinf(theta);
  float cx = cosf(XA),    sx = sinf(XA);
  float ym = (fabsf(st) + fabsf(ct)) * 31.5f;   // analytic _ymax

  float* d = derived + b * 8;
  d[0] = ct; d[1] = st; d[2] = cx; d[3] = sx;
  d[4] = 1.0f / SL; d[5] = 1.0f / SLZ; d[6] = ym; d[7] = 0.0f;
  xang_out[b] = XA;
}

// ---------------------------------------------------------------------------
// Cube maker: raw[b,i,j] = sum_k exp(-sqrt(xx^2+yy^2)/SL - |zz|/SLZ)
// ---------------------------------------------------------------------------
__global__ void cube_raw(const float* __restrict__ derived, float* __restrict__ raw) {
  int idx = blockIdx.x * blockDim.x + threadIdx.x;
  if (idx >= BATCH * 64 * 64) return;
  int j = idx & 63;
  int i = (idx >> 6) & 63;
  int b = idx >> 12;
  const float* d = derived + b * 8;
  float ct = d[0], st = d[1], cx = d[2], sx = d[3];
  float invSL = d[4], invSLZ = d[5], ym = d[6];

  const float inv = 1.0f / 31.5f;
  float xn = ((float)i - 31.5f) * inv;
  float yn = ((float)j - 31.5f) * inv;
  float xx = (xn * ct - yn * st) * 31.5f;
  float _y = (xn * st + yn * ct) * 31.5f;

  float c1 = _y * cx;                   // yy = c1 - zn * c2
  float c2 = sx * ym;
  float c3 = (_y / ym) * sx * 31.5f;    // zz = c3 + zn * c4
  float c4 = cx * 31.5f;
  float xx2 = xx * xx;

  float s = 0.0f;
#pragma unroll 8
  for (int k = 0; k < 64; ++k) {
    float zn = ((float)k - 31.5f) * inv;
    float yy = fmaf(-zn, c2, c1);
    float zz = fmaf(zn, c4, c3);
    float r  = sqrtf(fmaf(yy, yy, xx2));
    s += __expf(fmaf(-r, invSL, -fabsf(zz) * invSLZ));
  }
  raw[idx] = s;
}

// ---------------------------------------------------------------------------
// Per-sample min/max over 4096 values (one block per sample).
// ---------------------------------------------------------------------------
__global__ void minmax_kernel(const float* __restrict__ raw, float* __restrict__ mnmx) {
  __shared__ float smn[256];
  __shared__ float smx[256];
  int b = blockIdx.x;
  int t = threadIdx.x;
  float mn = INFINITY, mx = -INFINITY;
  for (int u = t; u < 4096; u += 256) {
    float v = raw[b * 4096 + u];
    mn = fminf(mn, v);
    mx = fmaxf(mx, v);
  }
  smn[t] = mn; smx[t] = mx;
  __syncthreads();
  for (int s = 128; s > 0; s >>= 1) {
    if (t < s) {
      smn[t] = fminf(smn[t], smn[t + s]);
      smx[t] = fmaxf(smx[t], smx[t + s]);
    }
    __syncthreads();
  }
  if (t == 0) { mnmx[b * 2] = smn[0]; mnmx[b * 2 + 1] = smx[0]; }
}

// ---------------------------------------------------------------------------
// Normalize + mask
// ---------------------------------------------------------------------------
__global__ void normalize_mask(const float* __restrict__ raw,
                               const float* __restrict__ mnmx,
                               const float* __restrict__ x1,
                               float* __restrict__ out) {
  int idx = blockIdx.x * blockDim.x + threadIdx.x;
  if (idx >= BATCH * 64 * 64) return;
  int b = idx >> 12;
  float mn = mnmx[b * 2], mx = mnmx[b * 2 + 1];
  float v = (raw[idx] - mn + 1e-10f) / (mx - mn + 1e-10f);
  out[idx] = (x1[idx] == 0.0f) ? 0.0f : v;
}

// ---------------------------------------------------------------------------
// Host-side launch
// ---------------------------------------------------------------------------
extern "C" void kernel_launch(void* const* d_in, const int* in_sizes, int n_in,
                              void* d_out, int out_size, void* d_ws, size_t ws_size,
                              hipStream_t stream) {
  const float* x1  = (const float*)d_in[0];
  const float* pos = (const float*)d_in[1];
  const float* w0  = (const float*)d_in[5];
  const float* b0  = (const float*)d_in[6];
  const float* w1  = (const float*)d_in[7];
  const float* b1  = (const float*)d_in[8];
  const float* w2  = (const float*)d_in[9];
  const float* b2  = (const float*)d_in[10];
  const float* w3  = (const float*)d_in[11];
  const float* b3  = (const float*)d_in[12];
  const float* wl1 = (const float*)d_in[13];
  const float* bl1 = (const float*)d_in[14];
  const float* wl2 = (const float*)d_in[15];
  const float* bl2 = (const float*)d_in[16];

  float* out = (float*)d_out;               // [64,1,64,64] bright
  float* xang_out = out + BATCH * 64 * 64;  // [64] X_ANGLE tail

  float* ws = (float*)d_ws;
  float* a0      = ws;                     // 64*16*32*32 = 1048576
  float* a1      = ws + 1048576;           // 64*32*16*16 = 524288
  float* a2      = ws + 1572864;           // 64*64*8*8   = 262144
  float* hflat   = ws + 1835008;           // 64*2048     = 131072
  float* h2      = ws + 1966080;           // 64*256      = 16384
  float* derived = ws + 1982464;           // 64*8        = 512
  float* raw     = ws + 1982976;           // 64*64*64    = 262144
  float* mnmx    = ws + 2245120;           // 128
  float* go      = ws + 2245248;           // reused GEMM out, up to 2097152

  // Encoder
  conv_relu_pool<1, 16, 64><<<4096, 256, 0, stream>>>(x1, w0, b0, a0);

  conv_wmma<16, 32, 32, 16><<<128, 256, 0, stream>>>(a0, w1, go);
  pool_bias_relu<32, 32><<<2048, 256, 0, stream>>>(go, b1, a1);

  conv_wmma<32, 64, 16, 16><<<64, 256, 0, stream>>>(a1, w2, go);
  pool_bias_relu<64, 16><<<1024, 256, 0, stream>>>(go, b2, a2);

  conv_wmma<64, 128, 8, 8><<<64, 256, 0, stream>>>(a2, w3, go);
  pool_bias_relu<128, 8><<<512, 256, 0, stream>>>(go, b3, hflat);

  // FC layers
  fc1_wmma<<<64, 32, 0, stream>>>(hflat, wl1, bl1, h2);
  fc2_derive<<<1, 64, 0, stream>>>(h2, wl2, bl2, pos, derived, xang_out);

  // Cube maker
  cube_raw<<<1024, 256, 0, stream>>>(derived, raw);
  minmax_kernel<<<64, 256, 0, stream>>>(raw, mnmx);
  normalize_mask<<<1024, 256, 0, stream>>>(raw, mnmx, x1, out);
}